// DynamicConv_65824668779127
// MI455X (gfx1250) — compile-verified
//
#include <hip/hip_runtime.h>
#include <math.h>

// ---------------------------------------------------------------------------
// MI455X (gfx1250) implementation of the GCNet-style ContextBlock head.
//
// Roofline: x is 512 MiB fp32; everything else is tiny. HBM floor at
// 23.3 TB/s is ~22us, so the spatial pool is built to read x from HBM once
// (split-softmax; chunk re-read for the weighted sum hits the 192MB L2).
// The dense 32x256x256 GEMM chain uses V_WMMA_F32_16X16X4_F32 (the fp32
// CDNA5 matrix op) so it stays numerically fp32 like the reference.
// ---------------------------------------------------------------------------

typedef __attribute__((ext_vector_type(2))) float v2f;
typedef __attribute__((ext_vector_type(8))) float v8f;

#define C_CH    256
#define HW_N    16384
#define BATCH   32
#define FEAT    256
#define CHUNKS  16                 // workgroups per batch for the spatial pool
#define POSW    (HW_N / CHUNKS)    // 1024 pixels per workgroup
#define PSTRIDE (C_CH + 2)         // per-chunk partial: [m, l, s[256]]

__device__ __forceinline__ float sigmoidf_(float v) {
  return 1.0f / (1.0f + __expf(-v));
}

// ---------------------------------------------------------------------------
// Kernel A: per-(batch,chunk) split-softmax spatial pool.
// Phase 1 (coalesced over pixels): logit_n = sum_c x[b,c,n]*w[c] + bias.
// Phase 2: chunk max + unnormalized exp into LDS.
// Phase 3 (per-channel rows, L2-resident re-read): s[c] = sum_n x[b,c,n]*p_n.
// Emits partials (m, sum p, s[256]) for exact cross-chunk combination.
// ---------------------------------------------------------------------------
__global__ __launch_bounds__(256) void k_spatial_pool(
    const float* __restrict__ x, const float* __restrict__ mask_w,
    const float* __restrict__ mask_b, float* __restrict__ part)
{
  __shared__ float s_w[C_CH];
  __shared__ float s_p[POSW];
  __shared__ float s_red[256];

  const int tid   = threadIdx.x;
  const int b     = blockIdx.x / CHUNKS;
  const int chunk = blockIdx.x % CHUNKS;
  const int n0    = chunk * POSW;
  const float* xb = x + (size_t)b * C_CH * HW_N;

  s_w[tid] = mask_w[tid];
  // Warm this thread's channel row for phase 3 (global_prefetch_b8).
  __builtin_prefetch(xb + (size_t)tid * HW_N + n0, 0, 0);
  __syncthreads();

  // ---- Phase 1: 4 pixels per thread, float4-coalesced over pixels ----
  const float bias = mask_b[0];
  float l0 = bias, l1 = bias, l2 = bias, l3 = bias;
  const float* p0 = xb + n0 + 4 * tid;
  #pragma unroll 4
  for (int c = 0; c < C_CH; ++c) {
    float4 v = *(const float4*)(p0 + (size_t)c * HW_N);
    float w = s_w[c];
    l0 = fmaf(v.x, w, l0);
    l1 = fmaf(v.y, w, l1);
    l2 = fmaf(v.z, w, l2);
    l3 = fmaf(v.w, w, l3);
  }

  // ---- Phase 2: chunk max, exp, partial sum of p ----
  s_red[tid] = fmaxf(fmaxf(l0, l1), fmaxf(l2, l3));
  __syncthreads();
  for (int off = 128; off > 0; off >>= 1) {
    if (tid < off) s_red[tid] = fmaxf(s_red[tid], s_red[tid + off]);
    __syncthreads();
  }
  const float m = s_red[0];
  __syncthreads();

  float e0 = __expf(l0 - m), e1 = __expf(l1 - m);
  float e2 = __expf(l2 - m), e3 = __expf(l3 - m);
  s_p[4 * tid + 0] = e0;
  s_p[4 * tid + 1] = e1;
  s_p[4 * tid + 2] = e2;
  s_p[4 * tid + 3] = e3;
  float lpart = (e0 + e1) + (e2 + e3);
  __syncthreads();

  // ---- Phase 3: thread == channel; row re-read hits L2; s_p is broadcast ----
  const float4* row = (const float4*)(xb + (size_t)tid * HW_N + n0);
  float acc = 0.0f;
  #pragma unroll 8
  for (int j = 0; j < POSW / 4; ++j) {
    float4 v = row[j];
    acc = fmaf(v.x, s_p[4 * j + 0], acc);
    acc = fmaf(v.y, s_p[4 * j + 1], acc);
    acc = fmaf(v.z, s_p[4 * j + 2], acc);
    acc = fmaf(v.w, s_p[4 * j + 3], acc);
  }

  // ---- Reduce sum(p) over the workgroup and emit partials ----
  s_red[tid] = lpart;
  __syncthreads();
  for (int off = 128; off > 0; off >>= 1) {
    if (tid < off) s_red[tid] += s_red[tid + off];
    __syncthreads();
  }
  float* o = part + (size_t)(b * CHUNKS + chunk) * PSTRIDE;
  if (tid == 0) { o[0] = m; o[1] = s_red[0]; }
  o[2 + tid] = acc;
}

// ---------------------------------------------------------------------------
// Kernel B: exact split-softmax combine -> context (B, C).
// ---------------------------------------------------------------------------
__global__ __launch_bounds__(256) void k_combine(
    const float* __restrict__ part, float* __restrict__ ctx)
{
  const int b = blockIdx.x, tid = threadIdx.x;
  float m = -INFINITY;
  for (int j = 0; j < CHUNKS; ++j)
    m = fmaxf(m, part[(size_t)(b * CHUNKS + j) * PSTRIDE]);
  float L = 0.0f, S = 0.0f;
  for (int j = 0; j < CHUNKS; ++j) {
    const float* pj = part + (size_t)(b * CHUNKS + j) * PSTRIDE;
    float w = __expf(pj[0] - m);
    L = fmaf(pj[1], w, L);
    S = fmaf(pj[2 + tid], w, S);
  }
  ctx[b * C_CH + tid] = S / L;
}

// ---------------------------------------------------------------------------
// One 16x16 output tile of  D += A(16xK) * B(Kx16)  with K=256 via
// V_WMMA_F32_16X16X4_F32. A comes from LDS (row stride 257 -> bank-conflict
// free), B streamed from global with a parametrized layout so the same code
// serves W[o*256+c] weights and the out_w[f, c, 2, 2] center tap (stride 25).
// Operand layout per ISA 7.12.2: A lane L holds (M=L%16, K=koff/koff+1),
// B lane L holds (K=koff/koff+1, N=L%16), D VGPR r = (M=r+8*(L/16), N=L%16).
// ---------------------------------------------------------------------------
__device__ __attribute__((always_inline)) inline v8f gemm_tile(
    const float* bufA, const float* __restrict__ W,
    int m0, int n0, int mrow, int koff,
    int rowStride, int kStride, int kBase)
{
  v8f acc = {};
  const float* arow = bufA + (m0 + mrow) * 257;
  const float* wrow = W + (size_t)(n0 + mrow) * rowStride + kBase;
  #pragma unroll 4
  for (int k = 0; k < C_CH; k += 4) {
    v2f a, bb;
    a.x = arow[k + koff];
    a.y = arow[k + koff + 1];
    const float* wp = wrow + (size_t)(k + koff) * kStride;
    bb.x = wp[0];
    bb.y = wp[kStride];
    acc = __builtin_amdgcn_wmma_f32_16x16x4_f32(
        false, a, false, bb, (short)0, acc, false, false);
  }
  return acc;
}

// ---------------------------------------------------------------------------
// Kernel C: whole dense chain in one workgroup (8 waves, 32 tiles of 16x16):
// h = ctx@cm1^T + b1 ; LayerNorm ; ReLU ; h2 = h@cm2^T + b2 ; sigmoid ;
// out = h2@out_w[:,:,2,2]^T + out_b ; att = sigmoid(h2@att_w^T + att_b) ;
// result = att * out  ->  d_out (B, F).
// ---------------------------------------------------------------------------
__global__ __launch_bounds__(256) void k_dense_chain(
    const float* __restrict__ ctx,
    const float* __restrict__ cm1_w, const float* __restrict__ cm1_b,
    const float* __restrict__ ln_g,  const float* __restrict__ ln_b,
    const float* __restrict__ cm2_w, const float* __restrict__ cm2_b,
    const float* __restrict__ out_w, const float* __restrict__ out_b,
    const float* __restrict__ att_w, const float* __restrict__ att_b,
    float* __restrict__ out)
{
  __shared__ float buf[BATCH * 257];
  __shared__ float s_mu[BATCH], s_rs[BATCH];

  const int tid  = threadIdx.x;
  const int lane = tid & 31;
  const int wave = tid >> 5;
  const int mrow = lane & 15;
  const int koff = (lane < 16) ? 0 : 2;
  const int dsel = (lane < 16) ? 0 : 8;

  for (int j = 0; j < BATCH; ++j) buf[j * 257 + tid] = ctx[j * C_CH + tid];
  __syncthreads();

  v8f acc[4];

  // ---- GEMM1: h = ctx @ cm1_w^T ----
  #pragma unroll
  for (int i = 0; i < 4; ++i) {
    int tile = wave + 8 * i, mt = tile & 1, nt = tile >> 1;
    acc[i] = gemm_tile(buf, cm1_w, mt * 16, nt * 16, mrow, koff, C_CH, 1, 0);
  }
  __syncthreads();   // all A reads done before overwriting buf
  #pragma unroll
  for (int i = 0; i < 4; ++i) {
    int tile = wave + 8 * i, mt = tile & 1, nt = tile >> 1;
    int n = nt * 16 + mrow;
    float bias = cm1_b[n];
    #pragma unroll
    for (int r = 0; r < 8; ++r)
      buf[(mt * 16 + r + dsel) * 257 + n] = acc[i][r] + bias;
  }
  __syncthreads();

  // ---- LayerNorm over C (biased var, eps=1e-5) + ReLU ----
  if (tid < BATCH) {
    float sum = 0.0f, sq = 0.0f;
    for (int c = 0; c < C_CH; ++c) {
      float v = buf[tid * 257 + c];
      sum += v;
      sq = fmaf(v, v, sq);
    }
    float mu  = sum * (1.0f / C_CH);
    float var = sq * (1.0f / C_CH) - mu * mu;
    s_mu[tid] = mu;
    s_rs[tid] = rsqrtf(var + 1e-5f);
  }
  __syncthreads();
  {
    float g = ln_g[tid], be = ln_b[tid];
    for (int j = 0; j < BATCH; ++j) {
      float v = buf[j * 257 + tid];
      v = fmaf((v - s_mu[j]) * s_rs[j], g, be);
      buf[j * 257 + tid] = fmaxf(v, 0.0f);
    }
  }
  __syncthreads();

  // ---- GEMM2: h2 = relu_h @ cm2_w^T, then sigmoid ----
  #pragma unroll
  for (int i = 0; i < 4; ++i) {
    int tile = wave + 8 * i, mt = tile & 1, nt = tile >> 1;
    acc[i] = gemm_tile(buf, cm2_w, mt * 16, nt * 16, mrow, koff, C_CH, 1, 0);
  }
  __syncthreads();
  #pragma unroll
  for (int i = 0; i < 4; ++i) {
    int tile = wave + 8 * i, mt = tile & 1, nt = tile >> 1;
    int n = nt * 16 + mrow;
    float bias = cm2_b[n];
    #pragma unroll
    for (int r = 0; r < 8; ++r)
      buf[(mt * 16 + r + dsel) * 257 + n] = sigmoidf_(acc[i][r] + bias);
  }
  __syncthreads();

  // ---- GEMM3 (5x5 center tap) + GEMM4 (att gate) + combine ----
  #pragma unroll
  for (int i = 0; i < 4; ++i) {
    int tile = wave + 8 * i, mt = tile & 1, nt = tile >> 1;
    v8f ao = gemm_tile(buf, out_w, mt * 16, nt * 16, mrow, koff,
                       C_CH * 25, 25, 2 * 5 + 2);
    v8f aa = gemm_tile(buf, att_w, mt * 16, nt * 16, mrow, koff, C_CH, 1, 0);
    int f = nt * 16 + mrow;
    float ob = out_b[f], ab = att_b[f];
    #pragma unroll
    for (int r = 0; r < 8; ++r) {
      int brow = mt * 16 + r + dsel;
      out[brow * FEAT + f] = sigmoidf_(aa[r] + ab) * (ao[r] + ob);
    }
  }
}

// ---------------------------------------------------------------------------
extern "C" void kernel_launch(void* const* d_in, const int* in_sizes, int n_in,
                              void* d_out, int out_size, void* d_ws,
                              size_t ws_size, hipStream_t stream)
{
  (void)in_sizes; (void)n_in; (void)out_size; (void)ws_size;
  const float* x      = (const float*)d_in[0];
  const float* mask_w = (const float*)d_in[1];
  const float* mask_b = (const float*)d_in[2];
  const float* cm1_w  = (const float*)d_in[3];
  const float* cm1_b  = (const float*)d_in[4];
  const float* ln_g   = (const float*)d_in[5];
  const float* ln_b   = (const float*)d_in[6];
  const float* cm2_w  = (const float*)d_in[7];
  const float* cm2_b  = (const float*)d_in[8];
  const float* out_w  = (const float*)d_in[9];
  const float* out_b  = (const float*)d_in[10];
  const float* att_w  = (const float*)d_in[11];
  const float* att_b  = (const float*)d_in[12];

  float* part = (float*)d_ws;                               // B*CHUNKS*258 f32
  float* ctx  = part + (size_t)BATCH * CHUNKS * PSTRIDE;    // B*C f32

  k_spatial_pool<<<BATCH * CHUNKS, 256, 0, stream>>>(x, mask_w, mask_b, part);
  k_combine<<<BATCH, 256, 0, stream>>>(part, ctx);
  k_dense_chain<<<1, 256, 0, stream>>>(ctx, cm1_w, cm1_b, ln_g, ln_b,
                                       cm2_w, cm2_b, out_w, out_b,
                                       att_w, att_b, (float*)d_out);
}